// SoftDTW_60516089201001
// MI455X (gfx1250) — compile-verified
//
#include <hip/hip_runtime.h>

typedef float v2f __attribute__((ext_vector_type(2)));
typedef float v8f __attribute__((ext_vector_type(8)));

#define GAMMA   1.0f
#define BIGF    1.0e10f
#define SEQN    512        // N == M == 512
#define NTILES  32         // 512 / 16

// One workgroup (512 threads = 16 wave32s) per batch.
// LDS: norms (4KB) + double-buffered D tile-strip (64KB) + 3 DP diagonals (~6KB).
__global__ __launch_bounds__(512)
void softdtw_fused_kernel(const float* __restrict__ xg,
                          const float* __restrict__ yg,
                          float* __restrict__ out)
{
    const int batch = blockIdx.x;
    const int tid   = threadIdx.x;
    const int lane  = tid & 31;          // wave32 lane
    const int wave  = tid >> 5;          // 0..15

    __shared__ float sx2[SEQN];
    __shared__ float sy2[SEQN];
    __shared__ float sD[2][NTILES * 256];   // two tile-anti-diagonal strips of D
    __shared__ float sr[3][SEQN + 8];       // rotating DP diagonals, i = 0..512

    const float* xb = xg + (size_t)batch * SEQN * 16;
    const float* yb = yg + (size_t)batch * SEQN * 16;

    // ---- row norms: thread tid handles row tid of both x and y ----
    {
        const float4* xr = (const float4*)(xb + tid * 16);
        const float4* yr = (const float4*)(yb + tid * 16);
        float ax = 0.f, ay = 0.f;
#pragma unroll
        for (int q = 0; q < 4; ++q) {
            float4 vx = xr[q], vy = yr[q];
            ax += vx.x * vx.x + vx.y * vx.y + vx.z * vx.z + vx.w * vx.w;
            ay += vy.x * vy.x + vy.y * vy.y + vy.z * vy.z + vy.w * vy.w;
        }
        sx2[tid] = ax;
        sy2[tid] = ay;
    }

    // ---- init DP diagonals: diag0 = [0, BIG...], diag1 = [BIG...] ----
    for (int idx = tid; idx <= SEQN; idx += 512) {
        sr[0][idx] = (idx == 0) ? 0.0f : BIGF;
        sr[1][idx] = BIGF;
    }
    __syncthreads();

    for (int k = 2; k <= 2 * SEQN; ++k) {
        // ---- every 16 cell-diagonals: build D tile-strip t = (k-2)/16 with WMMA ----
        if (((k - 2) & 15) == 0) {
            const int t     = (k - 2) >> 4;
            const int tiMin = (t > NTILES - 1) ? (t - (NTILES - 1)) : 0;
            const int tiMax = (t < NTILES - 1) ? t : (NTILES - 1);
            const int ntile = tiMax - tiMin + 1;
            const int mn    = lane & 15;          // row of A / col of B for this lane
            const int kb    = (lane >> 4) << 1;   // K sub-offset: lanes 16-31 hold K+2
            float* dst = sD[t & 1];
            for (int idx = wave; idx < ntile; idx += 16) {   // wave-uniform: EXEC all-1s
                const int ti = tiMin + idx;
                const int tj = t - ti;
                v8f acc = {0.f, 0.f, 0.f, 0.f, 0.f, 0.f, 0.f, 0.f};
                const float* xrow = xb + (ti * 16 + mn) * 16 + kb;
                const float* yrow = yb + (tj * 16 + mn) * 16 + kb;
#pragma unroll
                for (int kk = 0; kk < 4; ++kk) {  // K = 16 via 4 x (16x16x4 f32 WMMA)
                    v2f av = *(const v2f*)(xrow + kk * 4);
                    v2f bv = *(const v2f*)(yrow + kk * 4);
                    acc = __builtin_amdgcn_wmma_f32_16x16x4_f32(
                        false, av, false, bv, (short)0, acc, false, false);
                }
                // C/D layout: VGPR v, lane l -> M = v + 8*(l>=16), N = l%16
                const int mb = (lane >> 4) << 3;
#pragma unroll
                for (int v = 0; v < 8; ++v) {
                    const int mm = v + mb;
                    dst[(ti << 8) + (mm << 4) + mn] =
                        sx2[ti * 16 + mm] + sy2[tj * 16 + mn] - 2.0f * acc[v];
                }
            }
            __syncthreads();
        }

        // ---- DP step on cell anti-diagonal k: thread tid handles i = tid+1 ----
        float*       rNew = sr[k % 3];
        const float* r1   = sr[(k + 2) % 3];   // diag k-1
        const float* r2   = sr[(k + 1) % 3];   // diag k-2
        const int i = tid + 1;
        const int j = k - i;
        float val = BIGF;
        if (j >= 1 && j <= SEQN) {
            const int a = i - 1, c = j - 1;
            const float d = sD[((a >> 4) + (c >> 4)) & 1]
                              [((a >> 4) << 8) + ((a & 15) << 4) + (c & 15)];
            const float p0 = r2[i - 1];   // diagonal predecessor
            const float p1 = r1[i - 1];   // up
            const float p2 = r1[i];       // left
            const float m  = fminf(p0, fminf(p1, p2));
            const float s  = __expf((m - p0) * (1.0f / GAMMA))
                           + __expf((m - p1) * (1.0f / GAMMA))
                           + __expf((m - p2) * (1.0f / GAMMA));
            val = d + m - GAMMA * __logf(s);
        }
        rNew[i] = val;
        if (tid == 0) rNew[0] = BIGF;
        __syncthreads();
    }

    if (tid == 0) out[batch] = sr[(2 * SEQN) % 3][SEQN];
}

extern "C" void kernel_launch(void* const* d_in, const int* in_sizes, int n_in,
                              void* d_out, int out_size, void* d_ws, size_t ws_size,
                              hipStream_t stream) {
    const float* x = (const float*)d_in[0];   // (64, 512, 16) f32
    const float* y = (const float*)d_in[1];   // (64, 512, 16) f32
    float* out = (float*)d_out;               // (64,) f32
    (void)in_sizes; (void)n_in; (void)out_size; (void)d_ws; (void)ws_size;
    softdtw_fused_kernel<<<dim3(64), dim3(512), 0, stream>>>(x, y, out);
}